// GCN_59339268161961
// MI455X (gfx1250) — compile-verified
//
#include <hip/hip_runtime.h>
#include <hip/hip_bf16.h>

// ---------------------------------------------------------------------------
// 2-layer GCN for MI455X (gfx1250, wave32):
//   buf0 = spmm(adj, x)                      (edge-parallel f32 atomics, L2-resident)
//   buf1 = relu(buf0 @ W1 + b1)              (V_WMMA_F32_16X16X4_F32)
//   buf0 = spmm(adj, buf1)
//   out  = buf0 @ W2 + b2                    (V_WMMA_F32_16X16X4_F32)
//
// Roofline: ~2.5 GFLOP total, ~70MB unique traffic, intermediates L2-resident
// (192MB L2). Bound by SpMM gather + f32 atomics. Dense layers use the fp32
// WMMA path (exact reference numerics). W is staged to LDS with
// GLOBAL_LOAD_ASYNC_TO_LDS_B32 (ASYNCcnt) so the transpose-scatter staging
// has no VGPR round-trip and no per-element s_wait_loadcnt serialization.
// NOTE: M (=50000) is an exact multiple of 16, so output tiles are never
// partial and the epilogue stores are unguarded.
// ---------------------------------------------------------------------------

typedef __attribute__((ext_vector_type(2))) float v2f;
typedef __attribute__((ext_vector_type(8))) float v8f;

#if defined(__HIP_DEVICE_COMPILE__) && defined(__AMDGCN__)
#if !__has_builtin(__builtin_amdgcn_wmma_f32_16x16x4_f32)
#error "missing __builtin_amdgcn_wmma_f32_16x16x4_f32 for gfx1250"
#endif
#endif

// ---------------------------------------------------------------- zero fill
__global__ __launch_bounds__(256) void zero_f4(float4* __restrict__ p, int n4) {
    int i = blockIdx.x * 256 + threadIdx.x;
    if (i < n4) {
        float4 z; z.x = 0.f; z.y = 0.f; z.z = 0.f; z.w = 0.f;
        p[i] = z;
    }
}

// ------------------------------------------------------- SpMM (scatter-add)
// One wave32 per edge; each lane owns 4 of the 128 features (float4).
// x[col] gather is a coalesced 512B wave read; y[row] updates are coalesced
// hardware global_atomic_add_f32 (no-return -> STOREcnt path).
__global__ __launch_bounds__(256) void spmm_edge_atomic(
    const float* __restrict__ xf, const int* __restrict__ row,
    const int* __restrict__ col, const float* __restrict__ vals,
    float* __restrict__ y, int E)
{
    int e    = blockIdx.x * 8 + (threadIdx.x >> 5);
    int lane = threadIdx.x & 31;
    if (e >= E) return;
    int   r = row[e];
    int   c = col[e];
    float v = vals[e];
    const float4 xv = *(const float4*)(xf + (size_t)c * 128 + lane * 4);
    float* yp = y + (size_t)r * 128 + lane * 4;
    unsafeAtomicAdd(yp + 0, v * xv.x);
    unsafeAtomicAdd(yp + 1, v * xv.y);
    unsafeAtomicAdd(yp + 2, v * xv.z);
    unsafeAtomicAdd(yp + 3, v * xv.w);
}

// -------------------------------------------------- dense layer (fp32 WMMA)
// out[M x NN] = act(A[M x KDIM] @ W[KDIM x NN] + bias), one 16x16 tile per
// wave. W is staged through LDS in K-chunks of 32, stored transposed
// sB[n][k] with +2 pad so each lane's B fragment is a single 8B ds load.
// Staging uses async global->LDS scatter (ASYNCcnt), no data VGPRs.
// Requires M % 16 == 0 (no partial M tiles).
#define KC   32
#define LDSK (KC + 2)

template <bool RELU, int KDIM, int NN>
__global__ __launch_bounds__(256) void gemm_bias_wmma(
    const float* __restrict__ A, const float* __restrict__ W,
    const float* __restrict__ bias, float* __restrict__ out, int M)
{
    __shared__ float sB[NN][LDSK];             // NN=128 -> 17408 B, NN=64 -> 8704 B

    const int tid    = threadIdx.x;
    const int wave   = tid >> 5;
    const int lane   = tid & 31;
    constexpr int tilesN = NN >> 4;            // 8 or 4 (power of two)
    const int tilesM = M >> 4;                 // exact (M % 16 == 0)
    const int total  = tilesM * tilesN;

    const int  t     = blockIdx.x * 8 + wave;  // wave-uniform
    const bool valid = (t < total);            // wave-uniform: EXEC stays all-1s
    const int  tt    = valid ? t : 0;
    const int  mt    = tt / tilesN;            // compile-time shift
    const int  nt    = tt & (tilesN - 1);

    // A fragment addressing (16x4 f32): lane -> row M, koff = 2*(lane>>4)
    const int m     = (mt << 4) + (lane & 15);
    const int koff  = (lane >> 4) << 1;
    const int gn    = (nt << 4) + (lane & 15); // global column
    const float* Arow = A + (size_t)m * KDIM;

    // LDS byte address of sB base (low 32 bits of the generic pointer are the
    // wave-relative LDS offset per the gfx1250 aperture truncation rule).
    const unsigned ldsBase = (unsigned)(uintptr_t)(&sB[0][0]);

    v8f c = {};
    constexpr int PER_THREAD = (KC * NN) / 256; // 16 or 8, exact

    for (int k0 = 0; k0 < KDIM; k0 += KC) {
        // Async transpose-scatter stage: copies W[k0+kk][nn] from global memory into sB[nn][kk] in LDS.
        // Global side is coalesced (idx walks n fastest); LDS address comes
        // from a per-lane VGPR, so the scatter costs nothing extra.
#pragma unroll
        for (int r = 0; r < PER_THREAD; ++r) {
            int idx = r * 256 + tid;
            int kk  = idx / NN;                // compile-time shift
            int nn  = idx & (NN - 1);
            unsigned lds_addr = ldsBase + (unsigned)(nn * LDSK + kk) * 4u;
            unsigned goff     = (unsigned)((k0 + kk) * NN + nn) * 4u;
            asm volatile("global_load_async_to_lds_b32 %0, %1, %2"
                         :: "v"(lds_addr), "v"(goff), "s"(W)
                         : "memory");
        }
        asm volatile("s_wait_asynccnt 0x0" ::: "memory");
        __syncthreads();

#pragma unroll
        for (int kk = 0; kk < KC; kk += 4) {
            v2f a = *(const v2f*)(Arow + (k0 + kk + koff));
            v2f b = *(const v2f*)(&sB[gn][kk + koff]);
            // D = A(16x4) * B(4x16) + C  --> v_wmma_f32_16x16x4_f32
            c = __builtin_amdgcn_wmma_f32_16x16x4_f32(
                    false, a, false, b, (short)0, c, false, false);
        }
        __syncthreads();
    }

    if (valid) {
        // c[j]: row mbase+j, col gn. Tiles are never partial (M % 16 == 0):
        // 8 unguarded stores at constant byte offsets from one base pointer.
        const float bv    = bias[gn];
        const int   mbase = (mt << 4) + ((lane >> 4) << 3);
        float*      op    = out + (size_t)mbase * NN + gn;
#pragma unroll
        for (int j = 0; j < 8; ++j) {
            float v = c[j] + bv;
            if (RELU) v = fmaxf(v, 0.0f);
            op[j * NN] = v;
        }
    }
}

// ---------------------------------------------------------------------------
extern "C" void kernel_launch(void* const* d_in, const int* in_sizes, int n_in,
                              void* d_out, int out_size, void* d_ws, size_t ws_size,
                              hipStream_t stream) {
    const float* x    = (const float*)d_in[0];
    const int*   row  = (const int*)  d_in[1];
    const int*   col  = (const int*)  d_in[2];
    const float* vals = (const float*)d_in[3];
    const float* W1   = (const float*)d_in[4];
    const float* b1   = (const float*)d_in[5];
    const float* W2   = (const float*)d_in[6];
    const float* b2   = (const float*)d_in[7];
    float*       out  = (float*)d_out;

    constexpr int IN_DIM = 128;                // fixed by the reference model
    constexpr int HID    = 128;
    constexpr int OUT    = 64;
    const int N = in_sizes[0] / IN_DIM;        // 50000 (multiple of 16)
    const int E = in_sizes[1];                 // 640000

    float* buf0 = (float*)d_ws;                    // N x 128 (spmm accumulator)
    float* buf1 = buf0 + (size_t)N * HID;          // N x 128 (relu hidden)

    const int n4    = (N * HID) / 4;
    const int zgrid = (n4 + 255) / 256;
    const int egrid = (E + 7) / 8;                 // 8 edges (waves) per block

    // ---- layer 1 ----
    zero_f4<<<zgrid, 256, 0, stream>>>((float4*)buf0, n4);
    spmm_edge_atomic<<<egrid, 256, 0, stream>>>(x, row, col, vals, buf0, E);
    {
        int tiles = (N / 16) * (HID / 16);
        gemm_bias_wmma<true, IN_DIM, HID><<<(tiles + 7) / 8, 256, 0, stream>>>(
            buf0, W1, b1, buf1, N);
    }

    // ---- layer 2 ----
    zero_f4<<<zgrid, 256, 0, stream>>>((float4*)buf0, n4);
    spmm_edge_atomic<<<egrid, 256, 0, stream>>>(buf1, row, col, vals, buf0, E);
    {
        int tiles = (N / 16) * (OUT / 16);
        gemm_bias_wmma<false, HID, OUT><<<(tiles + 7) / 8, 256, 0, stream>>>(
            buf0, W2, b2, out, N);
    }
}